// GANLoss_79319456023015
// MI455X (gfx1250) — compile-verified
//
#include <hip/hip_runtime.h>

typedef float v2f __attribute__((ext_vector_type(2)));
typedef float v8f __attribute__((ext_vector_type(8)));

#define BSZ   64
#define TLEN  32
#define VOC   50257
#define NEL   (BSZ * TLEN)   // 2048
#define NTHR  256

__global__ __launch_bounds__(NTHR) void ganloss_wmma_kernel(
    const float* __restrict__ prob,
    const int*   __restrict__ samples,
    const float* __restrict__ reward,
    const float* __restrict__ c_phi_z,
    const float* __restrict__ c_phi_z_tilde,
    float*       __restrict__ out)
{
    __shared__ float newp[NEL];   // gathered probs, row-major (b, t)
    __shared__ float sums[BSZ];   // per-b row sums

    const int tid = threadIdx.x;

    // ---- Phase 1: 2048 scattered gathers, 8 per thread, all in flight ----
    #pragma unroll
    for (int j = 0; j < NEL / NTHR; ++j) {
        const int e = tid + j * NTHR;                 // e = b*T + t
        const int s = samples[e];
        newp[e] = prob[(size_t)e * (size_t)VOC + (size_t)s];
    }
    __syncthreads();

    // ---- Phase 2: 64x32 row-sum via V_WMMA_F32_16X16X4_F32 (wave 0) ----
    // D = A(16x4) * ones(4x16) + C  =>  D[m][n] = sum_k A[m][k], exact f32 FMA.
    if (tid < 32) {                                   // one full wave, EXEC all 1s
        const int lane = tid;
        const int mrow = lane & 15;                   // M within tile
        const int koff = (lane >> 4) << 1;            // lanes 16..31 carry K=2,3
        v2f bones; bones.x = 1.0f; bones.y = 1.0f;    // all-ones B (4x16)

        #pragma unroll
        for (int tile = 0; tile < 4; ++tile) {
            const int m0 = tile * 16;
            const float* rowp = &newp[(m0 + mrow) * TLEN];
            v8f c = {};
            #pragma unroll
            for (int kc = 0; kc < TLEN; kc += 4) {    // 8 K-chunks of 4
                v2f a;
                a.x = rowp[kc + koff];
                a.y = rowp[kc + koff + 1];
                // 8 args: (neg_a, A, neg_b, B, c_mod, C, reuse_a, reuse_b)
                c = __builtin_amdgcn_wmma_f32_16x16x4_f32(
                        false, a, false, bones, (short)0, c, false, false);
            }
            // D layout: VGPR j -> lanes 0-15: M=m0+j ; lanes 16-31: M=m0+8+j.
            // Every N column equals the row sum, so read column N=0 via
            // lane 0 (rows m0..m0+7) and lane 16 (rows m0+8..m0+15).
            if (mrow == 0) {
                const int base = m0 + ((lane >> 4) << 3);
                #pragma unroll
                for (int j = 0; j < 8; ++j) sums[base + j] = c[j];
            }
        }
    }
    __syncthreads();

    // ---- Phase 3: tiny affine combine + final reduction (64 elements) ----
    if (tid == 0) {
        float acc = 0.0f;
        for (int b = 0; b < BSZ; ++b) {
            const float t  = c_phi_z_tilde[b];
            const float l  = sums[b] * (reward[b] - t) + c_phi_z[b] - t;
            acc += l;
        }
        out[0] = -acc;
    }
}

extern "C" void kernel_launch(void* const* d_in, const int* in_sizes, int n_in,
                              void* d_out, int out_size, void* d_ws, size_t ws_size,
                              hipStream_t stream) {
    (void)in_sizes; (void)n_in; (void)out_size; (void)d_ws; (void)ws_size;
    const float* prob          = (const float*)d_in[0];
    const int*   samples       = (const int*)  d_in[1];
    const float* reward        = (const float*)d_in[2];
    const float* c_phi_z       = (const float*)d_in[3];
    const float* c_phi_z_tilde = (const float*)d_in[4];
    // d_in[5..7] are the python-int scalars (B, T, V); shapes are fixed by the
    // reference, so they are baked in as compile-time constants.
    ganloss_wmma_kernel<<<1, NTHR, 0, stream>>>(
        prob, samples, reward, c_phi_z, c_phi_z_tilde, (float*)d_out);
}